// VectorQuantizer_2508260901681
// MI455X (gfx1250) — compile-verified
//
#include <hip/hip_runtime.h>

typedef float v8f __attribute__((ext_vector_type(8)));
typedef float v2f __attribute__((ext_vector_type(2)));

#define KCODES 1024
#define DDIM   64
#define DECAYF 0.99f
#define EPSF   1e-5f

// ---------------------------------------------------------------------------
// Kernel 1: zero counts/dw scratch, compute per-code squared norms ||e_k||^2
// ---------------------------------------------------------------------------
__global__ __launch_bounds__(256) void vq_prep(const float* __restrict__ weight,
                                               float* __restrict__ enorm,
                                               float* __restrict__ counts,
                                               float* __restrict__ dw) {
  int tid = blockIdx.x * blockDim.x + threadIdx.x;
  if (tid < KCODES * DDIM) dw[tid] = 0.0f;
  if (tid < KCODES) {
    counts[tid] = 0.0f;
    float s = 0.0f;
#pragma unroll
    for (int d = 0; d < DDIM; ++d) {
      float w = weight[tid * DDIM + d];
      s += w * w;
    }
    enorm[tid] = s;
  }
}

// ---------------------------------------------------------------------------
// Kernel 2: WMMA distance GEMM + argmin + quantized gather + segment atomics.
// Block = 256 threads = 8 waves; each wave owns 16 rows.
// Double-buffered LDS weight tiles (padded stride 68 -> conflict-free b64
// reads in B layout); next tile's global load is issued before the barrier so
// it overlaps the 16-WMMA compute phase; enorm staged in LDS so the compute
// phase touches only DS. A is pre-scaled by -2 so the accumulator holds
// -2*x.e directly and distance-without-row-constant is en + c.
// ---------------------------------------------------------------------------
__global__ __launch_bounds__(256) void vq_argmin(const float* __restrict__ flat,
                                                 const float* __restrict__ weight,
                                                 const float* __restrict__ enorm,
                                                 float* __restrict__ counts,
                                                 float* __restrict__ dw,
                                                 float* __restrict__ qst,
                                                 float* __restrict__ indices,
                                                 int nrows) {
  __shared__ float btile[2][16 * 68];
  __shared__ float ebuf[2][16];

  const int lane = threadIdx.x & 31;
  const int wid  = threadIdx.x >> 5;   // wave in block: 0..7
  const int half = lane >> 4;          // selects K-pair {0,1} vs {2,3}
  const int m    = lane & 15;          // A: row-in-tile; B: col-in-tile
  const long rowbase = ((long)blockIdx.x * 8 + wid) * 16;

  // ---- load A slab (16 rows x 64) in WMMA A layout, pre-scaled by -2 ----
  v2f a[16];
  {
    const float* arow = flat + (rowbase + m) * DDIM + 2 * half;
#pragma unroll
    for (int s = 0; s < 16; ++s) {
      v2f t = *(const v2f*)(arow + 4 * s);
      a[s].x = -2.0f * t.x;
      a[s].y = -2.0f * t.y;
    }
  }

  // ---- software-pipelined weight tile staging ----
  const int r  = threadIdx.x >> 4;        // 0..15 code row within tile
  const int c4 = (threadIdx.x & 15) * 4;  // 0..60
  const float* wsrc = weight + (long)r * DDIM + c4;

  float4 pre  = *(const float4*)wsrc;                         // tile 0
  float  epre = (threadIdx.x < 16) ? enorm[threadIdx.x] : 0.0f;

  float bestval[8];
  int   bestidx[8];
#pragma unroll
  for (int v = 0; v < 8; ++v) { bestval[v] = 3.4e38f; bestidx[v] = 0; }

  const int NT = KCODES / 16;
  for (int ct = 0; ct < NT; ++ct) {
    const int p = ct & 1;
    // commit prefetched tile to LDS buffer p
    *(float4*)&btile[p][r * 68 + c4] = pre;
    if (threadIdx.x < 16) ebuf[p][threadIdx.x] = epre;
    // issue next tile's loads now: they complete during this tile's compute
    if (ct + 1 < NT) {
      pre = *(const float4*)(wsrc + (long)(ct + 1) * 16 * DDIM);
      if (threadIdx.x < 16) epre = enorm[(ct + 1) * 16 + threadIdx.x];
    }
    __syncthreads();  // single barrier/iter: safe with double buffering

    v8f c = {};  // accumulates -2 * x . e
    const float* brow = &btile[p][m * 68 + 2 * half];
#pragma unroll
    for (int s = 0; s < 16; ++s) {
      v2f b = *(const v2f*)(brow + 4 * s);
      c = __builtin_amdgcn_wmma_f32_16x16x4_f32(false, a[s], false, b,
                                                (short)0, c, false, false);
    }

    const float en = ebuf[p][m];  // ||e_n||^2 for this lane's column (from LDS)
#pragma unroll
    for (int v = 0; v < 8; ++v) {
      float d = en + c[v];  // ||e||^2 - 2 x.e  (row-constant ||x||^2 omitted)
      if (d < bestval[v]) { bestval[v] = d; bestidx[v] = ct * 16 + m; }
    }
  }

  // ---- cross-lane argmin within each 16-lane half (rows 0-7 / 8-15) ----
#pragma unroll
  for (int v = 0; v < 8; ++v) {
    float bv = bestval[v];
    int   bi = bestidx[v];
#pragma unroll
    for (int off = 1; off < 16; off <<= 1) {
      float ov = __shfl_xor(bv, off, 32);
      int   oi = __shfl_xor(bi, off, 32);
      if (ov < bv || (ov == bv && oi < bi)) { bv = ov; bi = oi; }  // first-index tie-break
    }
    bestidx[v] = bi;
  }

  // ---- outputs: quantized rows, indices, segment-sum atomics ----
  const int lane2 = lane * 2;
#pragma unroll
  for (int rr = 0; rr < 16; ++rr) {
    int idx = __shfl(bestidx[rr & 7], (rr < 8) ? 0 : 16, 32);  // broadcast row's code
    long row = rowbase + rr;
    v2f x = *(const v2f*)(flat + row * DDIM + lane2);
    v2f w = *(const v2f*)(weight + (long)idx * DDIM + lane2);
    v2f q;
    q.x = x.x + (w.x - x.x);  // literal straight-through arithmetic
    q.y = x.y + (w.y - x.y);
    *(v2f*)(qst + row * DDIM + lane2) = q;
    atomicAdd(&dw[idx * DDIM + lane2 + 0], x.x);
    atomicAdd(&dw[idx * DDIM + lane2 + 1], x.y);
    if (lane == 0) {
      atomicAdd(&counts[idx], 1.0f);
      indices[row] = (float)idx;
    }
  }
}

// ---------------------------------------------------------------------------
// Kernel 3: EMA updates. One block, 1024 threads (one per code).
// ---------------------------------------------------------------------------
__global__ __launch_bounds__(1024) void vq_finalize(const float* __restrict__ cluster_size,
                                                    const float* __restrict__ embed_avg,
                                                    const float* __restrict__ counts,
                                                    const float* __restrict__ dw,
                                                    float* __restrict__ out_weight,
                                                    float* __restrict__ out_cs,
                                                    float* __restrict__ out_ea) {
  __shared__ float red[KCODES];
  const int k = threadIdx.x;
  float craw = DECAYF * cluster_size[k] + (1.0f - DECAYF) * counts[k];
  red[k] = craw;
  __syncthreads();
#pragma unroll
  for (int off = KCODES / 2; off > 0; off >>= 1) {
    if (k < off) red[k] += red[k + off];
    __syncthreads();
  }
  const float n  = red[0];
  const float cs = (craw + EPSF) / (n + (float)KCODES * EPSF) * n;
  out_cs[k] = cs;
#pragma unroll 4
  for (int d = 0; d < DDIM; ++d) {
    float ea = DECAYF * embed_avg[k * DDIM + d] + (1.0f - DECAYF) * dw[k * DDIM + d];
    out_ea[k * DDIM + d]     = ea;
    out_weight[k * DDIM + d] = ea / cs;
  }
}

// ---------------------------------------------------------------------------
extern "C" void kernel_launch(void* const* d_in, const int* in_sizes, int n_in,
                              void* d_out, int out_size, void* d_ws, size_t ws_size,
                              hipStream_t stream) {
  const float* inputs       = (const float*)d_in[0];
  const float* weight       = (const float*)d_in[1];
  const float* cluster_size = (const float*)d_in[2];
  const float* embed_avg    = (const float*)d_in[3];

  const int total = in_sizes[0];      // B*C*H*W elements
  const int nrows = total / DDIM;     // 65536

  float* ws     = (float*)d_ws;
  float* enorm  = ws;                 // K
  float* counts = ws + KCODES;        // K
  float* dw     = ws + 2 * KCODES;    // K*D

  float* out        = (float*)d_out;
  float* qst        = out;                          // total
  float* indices    = qst + total;                  // nrows
  float* out_weight = indices + nrows;              // K*D
  float* out_cs     = out_weight + KCODES * DDIM;   // K
  float* out_ea     = out_cs + KCODES;              // K*D

  vq_prep<<<(KCODES * DDIM + 255) / 256, 256, 0, stream>>>(weight, enorm, counts, dw);

  int nblocks = nrows / 128;  // 8 waves x 16 rows per block (exact for 65536)
  vq_argmin<<<nblocks, 256, 0, stream>>>(inputs, weight, enorm, counts, dw,
                                         qst, indices, nrows);

  vq_finalize<<<1, KCODES, 0, stream>>>(cluster_size, embed_avg, counts, dw,
                                        out_weight, out_cs, out_ea);
}